// RecformerPooler_6803228197089
// MI455X (gfx1250) — compile-verified
//
#include <hip/hip_runtime.h>

typedef float v2f __attribute__((ext_vector_type(2)));
typedef float v8f __attribute__((ext_vector_type(8)));

#define B_N        16
#define S_N        2048
#define H_N        768
#define D_N        128
#define ATTR_MAX_N 8
#define ITEM_MAX_N 32
#define NSEG       (B_N * ATTR_MAX_N * ITEM_MAX_N)   /* 4096 */
#define ROWS_PER_BLOCK 32
#define LDS_STRIDE 772   /* 768 + 4 pad: 16x4 fragment reads hit 64 distinct banks */

// ---------------------------------------------------------------------------
// Zero the scratch accumulators (sums + counts) — required every launch since
// the harness does not re-poison between graph replays.
// ---------------------------------------------------------------------------
__global__ void recpool_zero_kernel(float* __restrict__ ws, int n) {
    int i = blockIdx.x * blockDim.x + threadIdx.x;
    if (i < n) ws[i] = 0.0f;
}

// ---------------------------------------------------------------------------
// Fused GEMM (fp32 WMMA 16x16x4) + bias + h store + segmented scatter-add.
// Block = 256 threads = 8 wave32; block owns 32 token rows; wave w owns
// columns [16w, 16w+16) and keeps TWO 16x16 accumulators (row panels m0 and
// m0+16), so every B fragment (W data, global) feeds 2 WMMAs.
// ---------------------------------------------------------------------------
__global__ __launch_bounds__(256)
void recpool_gemm_kernel(const float* __restrict__ hidden,
                         const int*   __restrict__ attr_ids,
                         const int*   __restrict__ item_ids,
                         const float* __restrict__ Wmat,
                         const float* __restrict__ bias,
                         float* __restrict__ h_out,
                         float* __restrict__ seg_sums,
                         float* __restrict__ seg_counts) {
    extern __shared__ float ldsA[];            // 32 * LDS_STRIDE floats (~96.5 KB)

    const int m0  = blockIdx.x * ROWS_PER_BLOCK;  // first token row of block
    const int tid = threadIdx.x;

    // ---- Stage A panel (32 x 768 fp32, contiguous in memory) into LDS ----
    const float4* src4 = (const float4*)(hidden + (size_t)m0 * H_N);
    #pragma unroll
    for (int it = 0; it < 24; ++it) {          // 6144 float4 / 256 threads
        int idx4 = it * 256 + tid;
        int idx  = idx4 * 4;
        int row  = idx / H_N;
        int col  = idx - row * H_N;
        float4 v = src4[idx4];
        float* dst = &ldsA[row * LDS_STRIDE + col];
        dst[0] = v.x; dst[1] = v.y; dst[2] = v.z; dst[3] = v.w;
    }
    __syncthreads();

    const int lane = tid & 31;
    const int wave = tid >> 5;
    const int n0   = wave * 16;
    const int n    = n0 + (lane & 15);         // output column for this lane
    const int kb   = (lane < 16) ? 0 : 2;      // K sub-offset per ISA layout
    const int arow = lane & 15;                // A row (within panel) for lane

    const float* wcol  = Wmat + n;             // column n of W, stride D_N
    const float* aptr0 = &ldsA[arow * LDS_STRIDE + kb];
    const float* aptr1 = aptr0 + 16 * LDS_STRIDE;

    v8f acc0 = {};                             // rows [m0,   m0+16)
    v8f acc1 = {};                             // rows [m0+16, m0+32)

    #pragma unroll 4
    for (int k0 = 0; k0 < H_N; k0 += 4) {
        // A fragments 16x4: reg0 = K=k0 (lo lanes) / k0+2 (hi), reg1 = +1
        v2f a0 = *(const v2f*)(aptr0 + k0);
        v2f a1 = *(const v2f*)(aptr1 + k0);
        // B fragment 4x16, row-striped across lanes; reused by both panels
        v2f bfrag;
        bfrag.x = wcol[(size_t)(k0 + kb)     * D_N];
        bfrag.y = wcol[(size_t)(k0 + kb + 1) * D_N];
        acc0 = __builtin_amdgcn_wmma_f32_16x16x4_f32(
            false, a0, false, bfrag, (short)0, acc0, false, false);
        acc1 = __builtin_amdgcn_wmma_f32_16x16x4_f32(
            false, a1, false, bfrag, (short)0, acc1, false, false);
    }

    // ---- Epilogue: bias, store h, scatter-add into segment sums ----
    const float bv = bias[n];

    #pragma unroll
    for (int p = 0; p < 2; ++p) {
        v8f acc = p ? acc1 : acc0;
        const int rbase = m0 + p * 16 + ((lane < 16) ? 0 : 8);  // hi lanes = M+8
        #pragma unroll
        for (int r = 0; r < 8; ++r) {
            float v = acc[r] + bv;
            int row = rbase + r;
            h_out[(size_t)row * D_N + n] = v;

            int a_id = attr_ids[row];
            int i_id = item_ids[row];
            if (a_id >= 1 && a_id <= ATTR_MAX_N && i_id >= 1 && i_id <= ITEM_MAX_N) {
                int bidx = row >> 11;                 // row / S_N
                int seg  = (bidx * ATTR_MAX_N + (a_id - 1)) * ITEM_MAX_N + (i_id - 1);
                atomicAdd(&seg_sums[(size_t)seg * D_N + n], v);
            }
        }
    }

    // One count increment per valid token row (block owns its rows exclusively).
    if (tid < ROWS_PER_BLOCK) {
        int row  = m0 + tid;
        int a_id = attr_ids[row];
        int i_id = item_ids[row];
        if (a_id >= 1 && a_id <= ATTR_MAX_N && i_id >= 1 && i_id <= ITEM_MAX_N) {
            int bidx = row >> 11;
            int seg  = (bidx * ATTR_MAX_N + (a_id - 1)) * ITEM_MAX_N + (i_id - 1);
            atomicAdd(&seg_counts[seg], 1.0f);
        }
    }
}

// ---------------------------------------------------------------------------
// pooled = sums / max(counts,1) ; empty = (counts == 0) as 0/1 float
// ---------------------------------------------------------------------------
__global__ void recpool_finalize_kernel(const float* __restrict__ seg_sums,
                                        const float* __restrict__ seg_counts,
                                        float* __restrict__ pooled,
                                        float* __restrict__ empty) {
    int seg = blockIdx.x;
    float c = seg_counts[seg];
    float div = (c == 0.0f) ? 1.0f : c;
    int d = threadIdx.x;
    pooled[(size_t)seg * D_N + d] = seg_sums[(size_t)seg * D_N + d] / div;
    if (d == 0) empty[seg] = (c == 0.0f) ? 1.0f : 0.0f;
}

// ---------------------------------------------------------------------------
extern "C" void kernel_launch(void* const* d_in, const int* in_sizes, int n_in,
                              void* d_out, int out_size, void* d_ws, size_t ws_size,
                              hipStream_t stream) {
    // inputs (setup_inputs order):
    //   0: attention_mask (int32, unused by reference math)
    //   1: hidden_states  (f32, B*S*H)
    //   2: attr_type_ids  (int32, B*S)
    //   3: item_position_ids (int32, B*S)
    //   4: W (f32, H*D)   5: b (f32, D)
    const float* hidden = (const float*)d_in[1];
    const int*   attr   = (const int*)d_in[2];
    const int*   item   = (const int*)d_in[3];
    const float* Wmat   = (const float*)d_in[4];
    const float* bias   = (const float*)d_in[5];

    // outputs concatenated: pooled [4096*128] | empty [4096] | h [32768*128]
    float* pooled = (float*)d_out;
    float* empty  = pooled + (size_t)NSEG * D_N;
    float* h_out  = empty + NSEG;

    // workspace: sums [4096*128] then counts [4096]
    float* seg_sums   = (float*)d_ws;
    float* seg_counts = seg_sums + (size_t)NSEG * D_N;
    const int ws_elems = NSEG * D_N + NSEG;

    recpool_zero_kernel<<<(ws_elems + 255) / 256, 256, 0, stream>>>(seg_sums, ws_elems);

    const int nrow_blocks = (B_N * S_N) / ROWS_PER_BLOCK;   // 1024
    const size_t lds_bytes = (size_t)ROWS_PER_BLOCK * LDS_STRIDE * sizeof(float);
    recpool_gemm_kernel<<<nrow_blocks, 256, lds_bytes, stream>>>(
        hidden, attr, item, Wmat, bias, h_out, seg_sums, seg_counts);

    recpool_finalize_kernel<<<NSEG, D_N, 0, stream>>>(seg_sums, seg_counts, pooled, empty);
}